// CGenGA_57604101373956
// MI455X (gfx1250) — compile-verified
//
#include <hip/hip_runtime.h>

// GCN diffusion step for MI455X (gfx1250, wave32).
//  - dense x@W via V_WMMA_F32_16X16X4_F32 (full f32, matches reference dtype)
//  - A tile (16 x K, <=16KB) staged into LDS per block via async-to-LDS
//    (GLOBAL_LOAD_ASYNC_TO_LDS_B128 + s_wait_asynccnt)
//  - W pre-swizzled into WMMA fragment order -> one global_load_b64 per B frag
//  - edge aggregation via f32 atomics (entire working set is L2-resident: 192MB L2)
//
// Workspace layout (assumes ws_size >= ~206MB):
//   X, Y, Z, Acc : N*128 f32 each
//   dinv : N f32 ; flag : N i32 ; temb : 64 f32 ; Wswz : 16384 f32

#define THREADS 256

typedef __attribute__((ext_vector_type(2))) float v2f;
typedef __attribute__((ext_vector_type(8))) float v8f;
typedef __attribute__((ext_vector_type(4))) int   v4i;

#if __has_builtin(__builtin_amdgcn_global_load_async_to_lds_b128) && \
    __has_builtin(__builtin_amdgcn_s_wait_asynccnt)
#define USE_ASYNC_LDS 1
typedef __attribute__((address_space(1))) v4i* gv4i_p;   // global int4*
typedef __attribute__((address_space(3))) v4i* lv4i_p;   // LDS int4*
#endif

__device__ __forceinline__ float d_silu(float x) {
  return x * (1.0f / (1.0f + __expf(-x)));
}

// ---------------- degree / dinv ----------------
__global__ void k_deg(const int* __restrict__ dst, float* __restrict__ deg, int E) {
  int e = blockIdx.x * blockDim.x + threadIdx.x;
  if (e < E) atomicAdd(deg + dst[e], 1.0f);
}

__global__ void k_dinv(float* __restrict__ deg, int N) {
  int i = blockIdx.x * blockDim.x + threadIdx.x;
  if (i < N) deg[i] = rsqrtf(deg[i] + 1.0f);
}

// ---------------- time embedding MLP (tiny; 1 block, 64 threads) ----------------
__global__ void k_time_mlp(const int* __restrict__ t,
                           const float* __restrict__ w1, const float* __restrict__ b1,
                           const float* __restrict__ w2, const float* __restrict__ b2,
                           float* __restrict__ temb) {
  __shared__ float emb[64];
  __shared__ float h[64];
  const int f = threadIdx.x;  // 0..63
  const float tv = (float)t[0];
  if (f < 32) {
    float freq = __expf(-logf(10000.0f) * (float)f / 31.0f);
    float a = tv * freq;
    emb[f] = sinf(a);
    emb[f + 32] = cosf(a);
  }
  __syncthreads();
  float s = b1[f];
  for (int k = 0; k < 64; ++k) s += emb[k] * w1[k * 64 + f];
  h[f] = d_silu(s);
  __syncthreads();
  float o = b2[f];
  for (int k = 0; k < 64; ++k) o += h[k] * w2[k * 64 + f];
  temb[f] = o;
}

// ---------------- label flags (ordered: anm=2 first, then norm=1 wins) ----------------
__global__ void k_setflag(const int* __restrict__ idx, int n, int val, int* __restrict__ flag) {
  int i = blockIdx.x * blockDim.x + threadIdx.x;
  if (i < n) flag[idx[i]] = val;
}

__global__ void k_build_xt(const float* __restrict__ noise, const float* __restrict__ temb,
                           const int* __restrict__ flag, const float* __restrict__ lab,
                           float* __restrict__ X, int total) {
  int i = blockIdx.x * blockDim.x + threadIdx.x;
  if (i >= total) return;
  int node = i >> 6, f = i & 63;
  float v = noise[i] + temb[f];
  int fl = flag[node];
  if (fl == 2)      v += lab[64 + f];   // label_emb[1]
  else if (fl == 1) v += lab[f];        // label_emb[0]
  X[i] = v;
}

// ---------------- weight pre-swizzle into WMMA B-fragment order ----------------
// Wswz[((t*(Nout/16)+ct)*32 + L)*2 + j] = W[(4t + 2*(L>=16) + j)*Nout + ct*16 + (L&15)]
__global__ void k_swizzle_w(const float* __restrict__ W, float* __restrict__ Wswz,
                            int Ktot, int Nout) {
  int i = blockIdx.x * blockDim.x + threadIdx.x;
  if (i >= Ktot * Nout) return;
  const int j = i & 1;
  const int L = (i >> 1) & 31;
  const int rest = i >> 6;          // t*(Nout/16)+ct
  const int nct = Nout >> 4;
  const int ct = rest % nct;
  const int t  = rest / nct;
  const int kk = 4 * t + ((L >> 4) << 1) + j;
  const int col = (ct << 4) + (L & 15);
  Wswz[i] = W[kk * Nout + col];
}

// ---------------- WMMA GEMM: Out[N,Nout] = [A0 | A1] @ W ----------------
// One wave per 16x16 output tile. A tile staged in LDS (async-to-LDS when
// available). B fragments come pre-swizzled from Wswz: one b64 load each.
// f32 WMMA 16x16x4 VGPR layout per CDNA5 ISA 7.12.2.
__global__ void k_gemm_wmma(const float* __restrict__ A0, int lda0, int K0,
                            const float* __restrict__ A1, int lda1, int K1,
                            const float* __restrict__ Wswz, int Nout,
                            float* __restrict__ Out, int lgcpr /* log2(Ktot/4) */) {
  extern __shared__ float sA[];  // 16 * Ktot floats (<=16KB)
  const int Ktot = K0 + K1;
  const size_t rowBase = (size_t)blockIdx.x * 16;  // N % 16 == 0 for this workload

  // ---- stage A tile: 16 rows x Ktot floats, in float4 chunks.
  // nchunk = 16*(Ktot/4) = 4*Ktot is a multiple of 256 -> uniform trip count,
  // EXEC stays all-ones for the async ops.
  const int cpr = 1 << lgcpr;                 // float4 chunks per row
  const int nchunk = cpr << 4;
  for (int i = threadIdx.x; i < nchunk; i += blockDim.x) {
    const int r = i >> lgcpr;
    const int kf = (i & (cpr - 1)) << 2;      // float offset within row
    const float* gp = (kf < K0) ? (A0 + (rowBase + r) * (size_t)lda0 + kf)
                                : (A1 + (rowBase + r) * (size_t)lda1 + (kf - K0));
    float* lp = sA + r * Ktot + kf;
#ifdef USE_ASYNC_LDS
    __builtin_amdgcn_global_load_async_to_lds_b128((gv4i_p)gp, (lv4i_p)lp, 0, 0);
#else
    *(float4*)lp = *(const float4*)gp;
#endif
  }
#ifdef USE_ASYNC_LDS
  __builtin_amdgcn_s_wait_asynccnt(0);
#endif
  __syncthreads();

  const int lane  = threadIdx.x & 31;
  const int wave  = threadIdx.x >> 5;
  const int nwave = blockDim.x >> 5;
  const int m     = lane & 15;
  const int khalf = (lane >> 4) << 1;   // 0 or 2
  const int nct   = Nout >> 4;

  for (int ct = wave; ct < nct; ct += nwave) {
    v8f acc = {0.f, 0.f, 0.f, 0.f, 0.f, 0.f, 0.f, 0.f};
    const float* ap = sA + m * Ktot + khalf;
    const float* bp = Wswz + ((size_t)ct * 32 + lane) * 2;
    const int bstride = nct * 64;  // floats per k-tile of fragments
    for (int t = 0; t < (Ktot >> 2); ++t) {
      v2f a = *(const v2f*)(ap + (t << 2));       // ds_load_b64
      v2f b = *(const v2f*)(bp + t * bstride);    // global_load_b64 (cache-hot)
      acc = __builtin_amdgcn_wmma_f32_16x16x4_f32(
          /*neg_a=*/false, a, /*neg_b=*/false, b,
          /*c_mod=*/(short)0, acc, /*reuse_a=*/false, /*reuse_b=*/false);
    }
    const int nOut = (ct << 4) + m;
    const size_t mBase = rowBase + ((lane >> 4) << 3);
#pragma unroll
    for (int v = 0; v < 8; ++v)
      Out[(mBase + v) * (size_t)Nout + nOut] = acc[v];
  }
}

// ---------------- edge scatter-add: Acc[dst] += Y[src] * dinv[src]*dinv[dst] ----------------
// one thread per (edge, 4-float chunk); F in {64,128}, shift = log2(F/4)
__global__ void k_edge_agg(const int* __restrict__ src, const int* __restrict__ dst,
                           const float* __restrict__ dinv, const float* __restrict__ Y,
                           float* __restrict__ Acc, long long total, int shift, int F) {
  long long idx = (long long)blockIdx.x * blockDim.x + threadIdx.x;
  if (idx >= total) return;
  const int c = (int)(idx & ((1 << shift) - 1));
  const long long e = idx >> shift;
  const int s = src[e], d = dst[e];
  const float nrm = dinv[s] * dinv[d];
  const float4 v = *((const float4*)(Y + (size_t)s * F) + c);
  float* ap = Acc + (size_t)d * F + (c << 2);
  atomicAdd(ap + 0, v.x * nrm);
  atomicAdd(ap + 1, v.y * nrm);
  atomicAdd(ap + 2, v.z * nrm);
  atomicAdd(ap + 3, v.w * nrm);
}

// ---------------- finalize: Out = silu(Acc + Y*dinv^2 + bias) ----------------
__global__ void k_finalize(const float* __restrict__ Y, const float* __restrict__ Acc,
                           const float* __restrict__ dinv, const float* __restrict__ bias,
                           float* __restrict__ Out, long long total, int shift) {
  long long i = (long long)blockIdx.x * blockDim.x + threadIdx.x;
  if (i >= total) return;
  const int node = (int)(i >> shift);
  const int f = (int)(i & ((1 << shift) - 1));
  const float di = dinv[node];
  const float v = Acc[i] + Y[i] * di * di + bias[f];
  Out[i] = d_silu(v);
}

extern "C" void kernel_launch(void* const* d_in, const int* in_sizes, int n_in,
                              void* d_out, int out_size, void* d_ws, size_t ws_size,
                              hipStream_t stream) {
  (void)n_in; (void)out_size; (void)ws_size;
  const int N = in_sizes[0] / 64;
  const int E = in_sizes[1] / 2;

  const float* noise = (const float*)d_in[0];
  const int*   eix   = (const int*)d_in[1];
  const int*   src   = eix;
  const int*   dst   = eix + E;
  const int*   t     = (const int*)d_in[2];
  const int*   anm   = (const int*)d_in[3]; const int n_anm = in_sizes[3];
  const int*   nor   = (const int*)d_in[4]; const int n_nor = in_sizes[4];
  const float* tw1 = (const float*)d_in[5];
  const float* tb1 = (const float*)d_in[6];
  const float* tw2 = (const float*)d_in[7];
  const float* tb2 = (const float*)d_in[8];
  const float* lab = (const float*)d_in[9];
  const float* w0 = (const float*)d_in[10]; const float* b0 = (const float*)d_in[11];
  const float* w1 = (const float*)d_in[12]; const float* b1 = (const float*)d_in[13];
  const float* w2 = (const float*)d_in[14]; const float* b2 = (const float*)d_in[15];
  const float* w3 = (const float*)d_in[16]; const float* b3 = (const float*)d_in[17];

  float* X    = (float*)d_ws;
  float* Y    = X + (size_t)N * 128;
  float* Z    = Y + (size_t)N * 128;
  float* Acc  = Z + (size_t)N * 128;
  float* dinv = Acc + (size_t)N * 128;
  int*   flag = (int*)(dinv + N);
  float* temb = (float*)(flag + N);
  float* Wswz = temb + 64;           // up to 256*64 floats (64KB)

  auto cdiv = [](long long a, long long b) { return (unsigned)((a + b - 1) / b); };
  const int rowBlocks = N / 16;  // N = 100000 -> 6250 exact

  // --- degree -> dinv ---
  (void)hipMemsetAsync(dinv, 0, (size_t)N * 4, stream);
  k_deg<<<cdiv(E, THREADS), THREADS, 0, stream>>>(dst, dinv, E);
  k_dinv<<<cdiv(N, THREADS), THREADS, 0, stream>>>(dinv, N);

  // --- time embedding MLP ---
  k_time_mlp<<<1, 64, 0, stream>>>(t, tw1, tb1, tw2, tb2, temb);

  // --- label flags (order matters: norm overrides anm) ---
  (void)hipMemsetAsync(flag, 0, (size_t)N * 4, stream);
  k_setflag<<<cdiv(n_anm, THREADS), THREADS, 0, stream>>>(anm, n_anm, 2, flag);
  k_setflag<<<cdiv(n_nor, THREADS), THREADS, 0, stream>>>(nor, n_nor, 1, flag);
  k_build_xt<<<cdiv((long long)N * 64, THREADS), THREADS, 0, stream>>>(
      noise, temb, flag, lab, X, N * 64);

  // --- conv0: X[N,64] @ w0[64,128] -> Y ; h0 = silu(...) in-place in Y ---
  k_swizzle_w<<<cdiv(64 * 128, THREADS), THREADS, 0, stream>>>(w0, Wswz, 64, 128);
  k_gemm_wmma<<<rowBlocks, 256, 16 * 64 * 4, stream>>>(X, 64, 64, nullptr, 0, 0, Wswz, 128, Y, 4);
  (void)hipMemsetAsync(Acc, 0, (size_t)N * 128 * 4, stream);
  k_edge_agg<<<cdiv((long long)E * 32, THREADS), THREADS, 0, stream>>>(
      src, dst, dinv, Y, Acc, (long long)E * 32, 5, 128);
  k_finalize<<<cdiv((long long)N * 128, THREADS), THREADS, 0, stream>>>(
      Y, Acc, dinv, b0, Y, (long long)N * 128, 7);

  // --- conv1: Y(h0)[N,128] @ w1[128,64] -> Z ; h1 in-place in Z ---
  k_swizzle_w<<<cdiv(128 * 64, THREADS), THREADS, 0, stream>>>(w1, Wswz, 128, 64);
  k_gemm_wmma<<<rowBlocks, 256, 16 * 128 * 4, stream>>>(Y, 128, 128, nullptr, 0, 0, Wswz, 64, Z, 5);
  (void)hipMemsetAsync(Acc, 0, (size_t)N * 64 * 4, stream);
  k_edge_agg<<<cdiv((long long)E * 16, THREADS), THREADS, 0, stream>>>(
      src, dst, dinv, Z, Acc, (long long)E * 16, 4, 64);
  k_finalize<<<cdiv((long long)N * 64, THREADS), THREADS, 0, stream>>>(
      Z, Acc, dinv, b1, Z, (long long)N * 64, 6);

  // --- conv2: Z(h1)[N,64] @ w2[64,128] -> X ; h2 in-place in X ---
  k_swizzle_w<<<cdiv(64 * 128, THREADS), THREADS, 0, stream>>>(w2, Wswz, 64, 128);
  k_gemm_wmma<<<rowBlocks, 256, 16 * 64 * 4, stream>>>(Z, 64, 64, nullptr, 0, 0, Wswz, 128, X, 4);
  (void)hipMemsetAsync(Acc, 0, (size_t)N * 128 * 4, stream);
  k_edge_agg<<<cdiv((long long)E * 32, THREADS), THREADS, 0, stream>>>(
      src, dst, dinv, X, Acc, (long long)E * 32, 5, 128);
  k_finalize<<<cdiv((long long)N * 128, THREADS), THREADS, 0, stream>>>(
      X, Acc, dinv, b2, X, (long long)N * 128, 7);

  // --- conv3: concat([h2=X, h0=Y])[N,256] @ w3[256,64] -> Z ; silu -> d_out ---
  k_swizzle_w<<<cdiv(256 * 64, THREADS), THREADS, 0, stream>>>(w3, Wswz, 256, 64);
  k_gemm_wmma<<<rowBlocks, 256, 16 * 256 * 4, stream>>>(X, 128, 128, Y, 128, 128, Wswz, 64, Z, 6);
  (void)hipMemsetAsync(Acc, 0, (size_t)N * 64 * 4, stream);
  k_edge_agg<<<cdiv((long long)E * 16, THREADS), THREADS, 0, stream>>>(
      src, dst, dinv, Z, Acc, (long long)E * 16, 4, 64);
  k_finalize<<<cdiv((long long)N * 64, THREADS), THREADS, 0, stream>>>(
      Z, Acc, dinv, b3, (float*)d_out, (long long)N * 64, 6);
}